// ADMG_RKHSDagma_27968827031915
// MI455X (gfx1250) — compile-verified
//
#include <hip/hip_runtime.h>

// Problem dims (match the reference).
#define DD 8
#define NN 1500

typedef __attribute__((ext_vector_type(2))) float v2f;
typedef __attribute__((ext_vector_type(8))) float v8f;

// CDNA5 async global->LDS copy path (ASYNCcnt-tracked), probed via
// __has_builtin so the file compiles on any toolchain; fallback is plain
// load/store staging. Host pass always takes the fallback branch (parsed
// only, never codegen'd).
#if defined(__HIP_DEVICE_COMPILE__) &&                                   \
    __has_builtin(__builtin_amdgcn_global_load_async_to_lds_b32) &&      \
    __has_builtin(__builtin_amdgcn_s_wait_asynccnt)
#define USE_ASYNC_LDS 1
#else
#define USE_ASYNC_LDS 0
#endif

// Round-2 diagnostic showed the builtin's params print as '__device__ int *'
// (addrspace(1)) — non-const, int-typed. LDS side is addrspace(3).
#define AS_G1(p) ((__attribute__((address_space(1))) int*)(p))
#define AS_L3(p) ((__attribute__((address_space(3))) int*)(p))

// ---------------------------------------------------------------------------
// Main streaming kernel: memory-bound (AI ~= 0.5 FLOP/byte, 648 MB streamed;
// ~28 us floor at 23.3 TB/s). Block = (row-tile of 8, j). 256 threads =
// 8 wave32s, one output row per wave. Coefficients live in LDS:
//   cg[l*D + a] = beta[j, a, l]   (48 KB)  -> dotted against grad_K2[j,i,:,:]
//   ca[l]       = alpha[j, l]     ( 6 KB)  -> dotted against K[j,i,:]
// grad_K2 rows (12000 f32) and K rows (1500 f32) are contiguous, so lanes
// stream them with coalesced global_load_b128.
// ---------------------------------------------------------------------------
__global__ __launch_bounds__(256) void rkhs_rows_kernel(
    const float* __restrict__ alpha,   // [D, N]
    const float* __restrict__ beta,    // [D, D, N]
    const float* __restrict__ Kmat,    // [D, N, N]
    const float* __restrict__ gradK2,  // [D, N, N, D]
    float* __restrict__ out)           // [N, D]
{
    __shared__ __align__(16) float cg[NN * DD];  // 48000 B
    __shared__ __align__(16) float ca[NN];       //  6000 B

    const int t = threadIdx.x;
    const int j = blockIdx.y;
    const float* __restrict__ betaj  = beta  + (size_t)j * DD * NN;
    const float* __restrict__ alphaj = alpha + (size_t)j * NN;

#if USE_ASYNC_LDS
    // Permuting gather straight into LDS: each lane supplies an independent
    // LDS destination, so the transpose beta[j,a,l] -> cg[l*8+a] is free and
    // no VGPR round-trip is needed. Tracked by ASYNCcnt.
    for (int idx = t; idx < NN * DD; idx += 256) {
        const int l = idx >> 3;   // idx / D
        const int a = idx & 7;    // idx % D
        __builtin_amdgcn_global_load_async_to_lds_b32(
            AS_G1(betaj + a * NN + l), AS_L3(cg + idx), 0, 0);
    }
    for (int idx = t; idx < NN; idx += 256) {
        __builtin_amdgcn_global_load_async_to_lds_b32(
            AS_G1(alphaj + idx), AS_L3(ca + idx), 0, 0);
    }
    __builtin_amdgcn_s_wait_asynccnt(0);   // async writes are not DScnt-covered
#else
    for (int idx = t; idx < NN * DD; idx += 256) {
        const int l = idx >> 3;
        const int a = idx & 7;
        cg[idx] = betaj[a * NN + l];
    }
    for (int idx = t; idx < NN; idx += 256) {
        ca[idx] = alphaj[idx];
    }
#endif
    __syncthreads();

    const int wave = t >> 5;
    const int lane = t & 31;
    const int i = blockIdx.x * 8 + wave;
    if (i >= NN) return;

    const float* __restrict__ rg = gradK2 + (size_t)(j * NN + i) * (NN * DD);
    const float* __restrict__ rk = Kmat   + (size_t)(j * NN + i) * NN;

    float ax = 0.f, ay = 0.f, az = 0.f, aw = 0.f;

    // grad_K2 part: 12000 floats per row, 128 floats per wave per step.
    for (int k = lane * 4; k < NN * DD; k += 128) {
        if (k + 1024 < NN * DD)
            __builtin_prefetch(rg + k + 1024, 0, 1);  // global_prefetch_b8
        const float4 g = *reinterpret_cast<const float4*>(rg + k);
        const float4 c = *reinterpret_cast<const float4*>(cg + k);
        ax = fmaf(g.x, c.x, ax);
        ay = fmaf(g.y, c.y, ay);
        az = fmaf(g.z, c.z, az);
        aw = fmaf(g.w, c.w, aw);
    }
    // K part: 1500 floats per row.
    for (int k = lane * 4; k < NN; k += 128) {
        const float4 g = *reinterpret_cast<const float4*>(rk + k);
        const float4 c = *reinterpret_cast<const float4*>(ca + k);
        ax = fmaf(g.x, c.x, ax);
        ay = fmaf(g.y, c.y, ay);
        az = fmaf(g.z, c.z, az);
        aw = fmaf(g.w, c.w, aw);
    }

    float s = (ax + ay) + (az + aw);
    // wave32 tree reduction
    #pragma unroll
    for (int off = 16; off > 0; off >>= 1)
        s += __shfl_down(s, off, 32);

    if (lane == 0) out[i * DD + j] = s;
}

// ---------------------------------------------------------------------------
// Sigma = L * L^T + 1e-6 * I via one wave of v_wmma_f32_16x16x4_f32.
// 8x8x8 matmul zero-padded into the 16x16x4 shape, two K-steps (k0 = 0, 4).
// Ls is padded to 16 rows of zeros so fragment loads are unconditional
// (no exec save/restore around the ds_loads).
// Fragment layout (CDNA5 ISA):
//   A (16x4, 2 VGPRs): lane L<16 -> row m=L, K = k0,k0+1 ; L>=16 -> K = k0+2,k0+3
//   B (4x16, 2 VGPRs): lane L<16 -> col n=L, rows K = k0,k0+1 ; L>=16 -> k0+2,k0+3
//   C/D (16x16, 8 VGPRs): vgpr v, lane L -> row M = v + (L<16 ? 0 : 8), col = L&15
// Since Sigma is symmetric, per-lane A and B fragment values are identical.
// ---------------------------------------------------------------------------
__global__ __launch_bounds__(32) void sigma_wmma_kernel(
    const float* __restrict__ Lmat,  // [D, D]
    float* __restrict__ sigma)       // [D, D]
{
    __shared__ __align__(16) float Ls[16 * DD];   // rows 8..15 zero padding
    const int lane = threadIdx.x;
    for (int idx = lane; idx < 16 * DD; idx += 32)
        Ls[idx] = (idx < DD * DD) ? Lmat[idx] : 0.f;
    __syncthreads();

    v8f acc = {};
    const int m  = lane & 15;           // row (A) / col (B)
    const int kh = (lane >> 4) << 1;    // +0 for lanes 0-15, +2 for lanes 16-31

    #pragma unroll
    for (int k0 = 0; k0 < DD; k0 += 4) {
        const int kk = k0 + kh;         // 0,2,4,6 -> kk+1 < 8 always
        v2f a;
        a.x = Ls[m * DD + kk];          // A[m][kk]   = L[m][kk]
        a.y = Ls[m * DD + kk + 1];      // A[m][kk+1]
        // B[kk][n] = (L^T)[kk][n] = L[n][kk]; n == m here -> same values as A.
        v2f b = a;
        acc = __builtin_amdgcn_wmma_f32_16x16x4_f32(
            /*neg_a=*/false, a, /*neg_b=*/false, b,
            /*c_mod=*/(short)0, acc, /*reuse_a=*/false, /*reuse_b=*/false);
    }

    // Rows 0..7 of D live in vgprs 0..7 on lanes 0..15; keep cols 0..7.
    if (lane < DD) {
        #pragma unroll
        for (int v = 0; v < DD; ++v)
            sigma[v * DD + lane] = acc[v] + (v == lane ? 1e-6f : 0.f);
    }
}

extern "C" void kernel_launch(void* const* d_in, const int* in_sizes, int n_in,
                              void* d_out, int out_size, void* d_ws, size_t ws_size,
                              hipStream_t stream) {
    (void)in_sizes; (void)n_in; (void)d_ws; (void)ws_size; (void)out_size;

    const float* alpha  = (const float*)d_in[0];  // [D, N]
    const float* beta   = (const float*)d_in[1];  // [D, D, N]
    const float* Lmat   = (const float*)d_in[2];  // [D, D]
    const float* Kmat   = (const float*)d_in[3];  // [D, N, N]
    const float* gradK2 = (const float*)d_in[4];  // [D, N, N, D]

    float* out   = (float*)d_out;               // [N, D] flat first
    float* sigma = out + (size_t)NN * DD;       // then [D, D]

    dim3 grid((NN + 7) / 8, DD, 1);
    rkhs_rows_kernel<<<grid, 256, 0, stream>>>(alpha, beta, Kmat, gradK2, out);
    sigma_wmma_kernel<<<1, 32, 0, stream>>>(Lmat, sigma);
}